// CMCV3Loss_2508260900945
// MI455X (gfx1250) — compile-verified
//
#include <hip/hip_runtime.h>
#include <math.h>

#define INV_TEMP 14.285714285714286f
#define NEG_BIG  -1.0e30f

typedef __attribute__((ext_vector_type(2))) float v2f;
typedef __attribute__((ext_vector_type(8))) float v8f;

__device__ __forceinline__ float grp16_sum(float v) {
#pragma unroll
    for (int off = 1; off < 16; off <<= 1) v += __shfl_xor(v, off, 32);
    return v;
}

__device__ __forceinline__ void lse_update(float v, float& m, float& s) {
    if (v > m) { s = s * __expf(m - v) + 1.0f; m = v; }
    else       { s += __expf(v - m); }
}

// ---------------------------------------------------------------- init
__global__ void zero_out_kernel(float* out, int n) {
    int i = blockIdx.x * blockDim.x + threadIdx.x;
    if (i < n) out[i] = 0.0f;
}

// ---------------------------------------------------------------- norms
// One wave per row (128 floats = 32 lanes x float4).
// invf: 1/||row[0:128]||, invs: 1/||row[0:64]||, invp: 1/||row[64:128]||
__global__ void norms_kernel(const float* __restrict__ x0, const float* __restrict__ x1,
                             const float* __restrict__ x2, const float* __restrict__ x3,
                             float* __restrict__ invf, float* __restrict__ invs,
                             float* __restrict__ invp) {
    int input = blockIdx.y;
    const float* x = (input == 0) ? x0 : (input == 1) ? x1 : (input == 2) ? x2 : x3;
    int wave = threadIdx.x >> 5;
    int lane = threadIdx.x & 31;
    int row = blockIdx.x * 8 + wave;                // gridDim.x = 512 -> 4096 rows
    float4 v = *(const float4*)(x + row * 128 + lane * 4);
    float sq = v.x * v.x + v.y * v.y + v.z * v.z + v.w * v.w;
    float halfsum = grp16_sum(sq);                  // lanes 0-15: sh, 16-31: pv
    float full = halfsum + __shfl_xor(halfsum, 16, 32);
    if (lane == 0) {
        invf[input * 4096 + row] = 1.0f / fmaxf(sqrtf(full), 1e-8f);
        invs[input * 4096 + row] = 1.0f / fmaxf(sqrtf(halfsum), 1e-8f);
    }
    if (lane == 16)
        invp[input * 4096 + row] = 1.0f / fmaxf(sqrtf(halfsum), 1e-8f);
}

// ---------------------------------------------------------------- temporal
// Per wave: 16-row stripe of the 4096x4096 cosine-sim matrix (K=128),
// streaming column tiles through v_wmma_f32_16x16x4_f32 with online LSE.
// B operand is double-buffered in 16-chunk register groups so loads for the
// next group/tile overlap the current WMMA chain instead of stalling per-load.
__global__ void __launch_bounds__(256)
temporal_kernel(const float* __restrict__ x0, const float* __restrict__ x1,
                const float* __restrict__ x2, const float* __restrict__ x3,
                const float* __restrict__ invf_all,
                float* __restrict__ out, float weight) {
    int input = blockIdx.y;
    const float* __restrict__ x = (input == 0) ? x0 : (input == 1) ? x1 : (input == 2) ? x2 : x3;
    const float* __restrict__ invf = invf_all + input * 4096;

    int wave = threadIdx.x >> 5;
    int lane = threadIdx.x & 31;
    int l15 = lane & 15;
    int half = lane >> 4;
    int tile = blockIdx.x * 8 + wave;               // gridDim.x = 32 -> 256 tiles
    int rowbase = tile * 16;

    // A operand resident: 16 rows x K=128 -> 32 chunks of float2 per lane
    const float* arow = x + (rowbase + l15) * 128 + half * 2;
    v2f a[32];
#pragma unroll
    for (int k = 0; k < 32; ++k) a[k] = *(const v2f*)(arow + k * 4);

    float invr_t[8];
#pragma unroll
    for (int e = 0; e < 8; ++e) invr_t[e] = invf[rowbase + e + half * 8] * INV_TEMP;

    float m[8], s[8], pos[8];
#pragma unroll
    for (int e = 0; e < 8; ++e) { m[e] = -INFINITY; s[e] = 0.0f; pos[e] = 0.0f; }

    const float* xbase = x + l15 * 128 + half * 2;  // + col*128 (+ kgroup*64)
    v2f bA[16], bB[16];

    // prologue: first half of tile 0
#pragma unroll
    for (int k = 0; k < 16; ++k) bA[k] = *(const v2f*)(xbase + k * 4);

    for (int ct = 0; ct < 256; ++ct) {
        int colbase = ct * 16;
        const float* brow = xbase + colbase * 128;

        // issue loads for second K-half of this tile (overlaps chain on bA)
#pragma unroll
        for (int k = 0; k < 16; ++k) bB[k] = *(const v2f*)(brow + 64 + k * 4);

        v8f acc = {0.f, 0.f, 0.f, 0.f, 0.f, 0.f, 0.f, 0.f};
#pragma unroll
        for (int k = 0; k < 16; ++k)
            acc = __builtin_amdgcn_wmma_f32_16x16x4_f32(false, a[k], false, bA[k],
                                                        (short)0, acc, false, false);

        // issue loads for first K-half of the NEXT tile (overlaps chain on bB)
        const float* nrow = xbase + (((ct + 1) & 255) * 16) * 128;
#pragma unroll
        for (int k = 0; k < 16; ++k) bA[k] = *(const v2f*)(nrow + k * 4);

#pragma unroll
        for (int k = 0; k < 16; ++k)
            acc = __builtin_amdgcn_wmma_f32_16x16x4_f32(false, a[16 + k], false, bB[k],
                                                        (short)0, acc, false, false);

        float invc = invf[colbase + l15];
        if (ct == tile) {
            // diagonal + positive columns both live in this tile
            int col = colbase + l15;
#pragma unroll
            for (int e = 0; e < 8; ++e) {
                int row = rowbase + e + half * 8;
                float v = acc[e] * invr_t[e] * invc;
                int poscol = (row & ~7) + (((row & 7) == 0) ? 1 : 0);
                if (col == poscol) pos[e] += v;
                if (col == row) v = NEG_BIG;        // exclude self-diagonal
                lse_update(v, m[e], s[e]);
            }
        } else {
#pragma unroll
            for (int e = 0; e < 8; ++e)
                lse_update(acc[e] * invr_t[e] * invc, m[e], s[e]);
        }
    }

    float total = 0.0f;
#pragma unroll
    for (int e = 0; e < 8; ++e) {
        float mm = m[e], ss = s[e];
#pragma unroll
        for (int off = 1; off < 16; off <<= 1) {
            float m2 = __shfl_xor(mm, off, 32);
            float s2 = __shfl_xor(ss, off, 32);
            float mx = fmaxf(mm, m2);
            ss = ss * __expf(mm - mx) + s2 * __expf(m2 - mx);
            mm = mx;
        }
        float p = grp16_sum(pos[e]);
        total += (mm + __logf(ss)) - p;
    }
    if (l15 == 0) atomicAdd(out, weight * total);
}

// ---------------------------------------------------------------- contrastive
// Per position p (8): 1024x1024 sim over concat(viewA, viewB) slices (K=64).
// Row n<512 -> xa[(n*8+p)], n>=512 -> xb[((n-512)*8+p)]; slice offset coloff.
// Diagonal lives in tile ct==rt; positive partner in tile (rt+32)&63.
__global__ void __launch_bounds__(256)
contrastive_kernel(const float* __restrict__ xa, const float* __restrict__ xb,
                   const float* __restrict__ inva, const float* __restrict__ invb,
                   int coloff, float* __restrict__ out, float weight) {
    int wave = threadIdx.x >> 5;
    int lane = threadIdx.x & 31;
    int l15 = lane & 15;
    int half = lane >> 4;
    int t = blockIdx.x * 8 + wave;                  // gridDim.x = 64 -> 512 waves
    int p = t >> 6;                                 // temporal position 0..7
    int rt = t & 63;
    int rowbase = rt * 16;
    int postile = (rt + 32) & 63;

    bool rowInA = rowbase < 512;
    const float* rptr = rowInA ? xa : xb;
    const float* rinv = rowInA ? inva : invb;
    int rb = rowInA ? rowbase : rowbase - 512;

    const float* arow = rptr + ((rb + l15) * 8 + p) * 128 + coloff + half * 2;
    v2f a[16];
#pragma unroll
    for (int k = 0; k < 16; ++k) a[k] = *(const v2f*)(arow + k * 4);

    float invr_t[8];
#pragma unroll
    for (int e = 0; e < 8; ++e) invr_t[e] = rinv[(rb + e + half * 8) * 8 + p] * INV_TEMP;

    float m[8], s[8], pos[8];
#pragma unroll
    for (int e = 0; e < 8; ++e) { m[e] = -INFINITY; s[e] = 0.0f; pos[e] = 0.0f; }

    // per-tile B base pointer (column-half dependent)
    auto bbase = [&](int ct) -> const float* {
        int colbase = ct * 16;
        bool colInA = colbase < 512;
        const float* cptr = colInA ? xa : xb;
        int cb = colInA ? colbase : colbase - 512;
        return cptr + ((cb + l15) * 8 + p) * 128 + coloff + half * 2;
    };

    v2f bA[8], bB[8];
    {
        const float* brow = bbase(0);
#pragma unroll
        for (int k = 0; k < 8; ++k) bA[k] = *(const v2f*)(brow + k * 4);
    }

    for (int ct = 0; ct < 64; ++ct) {
        const float* brow = bbase(ct);
#pragma unroll
        for (int k = 0; k < 8; ++k) bB[k] = *(const v2f*)(brow + 32 + k * 4);

        v8f acc = {0.f, 0.f, 0.f, 0.f, 0.f, 0.f, 0.f, 0.f};
#pragma unroll
        for (int k = 0; k < 8; ++k)
            acc = __builtin_amdgcn_wmma_f32_16x16x4_f32(false, a[k], false, bA[k],
                                                        (short)0, acc, false, false);

        const float* nrow = bbase((ct + 1) & 63);
#pragma unroll
        for (int k = 0; k < 8; ++k) bA[k] = *(const v2f*)(nrow + k * 4);

#pragma unroll
        for (int k = 0; k < 8; ++k)
            acc = __builtin_amdgcn_wmma_f32_16x16x4_f32(false, a[8 + k], false, bB[k],
                                                        (short)0, acc, false, false);

        int colbase = ct * 16;
        bool colInA = colbase < 512;
        const float* cinv = colInA ? inva : invb;
        int cb = colInA ? colbase : colbase - 512;
        float invc = cinv[(cb + l15) * 8 + p];

        if (ct == rt) {
            int col = colbase + l15;
#pragma unroll
            for (int e = 0; e < 8; ++e) {
                int n = rowbase + e + half * 8;
                float v = acc[e] * invr_t[e] * invc;
                if (col == n) v = NEG_BIG;          // exclude self-diagonal
                lse_update(v, m[e], s[e]);
            }
        } else if (ct == postile) {
            int col = colbase + l15;
#pragma unroll
            for (int e = 0; e < 8; ++e) {
                int n = rowbase + e + half * 8;
                float v = acc[e] * invr_t[e] * invc;
                int poscol = (n + 512) & 1023;      // NT-Xent partner
                if (col == poscol) pos[e] += v;
                lse_update(v, m[e], s[e]);
            }
        } else {
#pragma unroll
            for (int e = 0; e < 8; ++e)
                lse_update(acc[e] * invr_t[e] * invc, m[e], s[e]);
        }
    }

    float total = 0.0f;
#pragma unroll
    for (int e = 0; e < 8; ++e) {
        float mm = m[e], ss = s[e];
#pragma unroll
        for (int off = 1; off < 16; off <<= 1) {
            float m2 = __shfl_xor(mm, off, 32);
            float s2 = __shfl_xor(ss, off, 32);
            float mx = fmaxf(mm, m2);
            ss = ss * __expf(mm - mx) + s2 * __expf(m2 - mx);
            mm = mx;
        }
        float p2 = grp16_sum(pos[e]);
        total += (mm + __logf(ss)) - p2;
    }
    if (l15 == 0) atomicAdd(out, weight * total);
}

// ---------------------------------------------------------------- ortho
// mean(relu(cos(a_row, b_row))) over 4096 rows, 64-dim slices.
__global__ void ortho_kernel(const float* __restrict__ xa, int offa, const float* __restrict__ inva,
                             const float* __restrict__ xb, int offb, const float* __restrict__ invb,
                             float* __restrict__ out, float weight) {
    int r = blockIdx.x * blockDim.x + threadIdx.x;  // 4096 rows
    const float* pa = xa + r * 128 + offa;
    const float* pb = xb + r * 128 + offb;
    float dot = 0.0f;
#pragma unroll
    for (int k = 0; k < 16; ++k) {
        float4 va = *(const float4*)(pa + k * 4);
        float4 vb = *(const float4*)(pb + k * 4);
        dot += va.x * vb.x + va.y * vb.y + va.z * vb.z + va.w * vb.w;
    }
    float c = fmaxf(dot * inva[r] * invb[r], 0.0f);
#pragma unroll
    for (int off = 1; off < 32; off <<= 1) c += __shfl_xor(c, off, 32);
    if ((threadIdx.x & 31) == 0) atomicAdd(out, weight * c);
}

// ---------------------------------------------------------------- launch
extern "C" void kernel_launch(void* const* d_in, const int* in_sizes, int n_in,
                              void* d_out, int out_size, void* d_ws, size_t ws_size,
                              hipStream_t stream) {
    const float* x0 = (const float*)d_in[0];   // mod1_v1
    const float* x1 = (const float*)d_in[1];   // mod1_v2
    const float* x2 = (const float*)d_in[2];   // mod2_v1
    const float* x3 = (const float*)d_in[3];   // mod2_v2
    float* out = (float*)d_out;

    float* ws   = (float*)d_ws;
    float* invf = ws;                  // 4 * 4096
    float* invs = ws + 4 * 4096;       // 4 * 4096
    float* invp = ws + 8 * 4096;       // 4 * 4096

    zero_out_kernel<<<1, 32, 0, stream>>>(out, out_size);

    norms_kernel<<<dim3(512, 4), 256, 0, stream>>>(x0, x1, x2, x3, invf, invs, invp);

    // temporal: all four inputs, mean over 4096 rows each
    temporal_kernel<<<dim3(32, 4), 256, 0, stream>>>(x0, x1, x2, x3, invf, out, 1.0f / 4096.0f);

    const float wc = 1.0f / 8192.0f;   // mean over s*2b rows
    // shared (first-half slices): (mod1_v1, mod2_v1) and (mod1_v2, mod2_v2)
    contrastive_kernel<<<64, 256, 0, stream>>>(x0, x2, invs + 0 * 4096, invs + 2 * 4096, 0,  out, wc);
    contrastive_kernel<<<64, 256, 0, stream>>>(x1, x3, invs + 1 * 4096, invs + 3 * 4096, 0,  out, wc);
    // private (second-half slices): (mod1_v1, mod1_v2) and (mod2_v1, mod2_v2)
    contrastive_kernel<<<64, 256, 0, stream>>>(x0, x1, invp + 0 * 4096, invp + 1 * 4096, 64, out, wc);
    contrastive_kernel<<<64, 256, 0, stream>>>(x2, x3, invp + 2 * 4096, invp + 3 * 4096, 64, out, wc);

    const float wo = 1.0f / 4096.0f;
    // shared-vs-private per modality per view
    ortho_kernel<<<16, 256, 0, stream>>>(x0, 0,  invs + 0 * 4096, x0, 64, invp + 0 * 4096, out, wo);
    ortho_kernel<<<16, 256, 0, stream>>>(x2, 0,  invs + 2 * 4096, x2, 64, invp + 2 * 4096, out, wo);
    ortho_kernel<<<16, 256, 0, stream>>>(x1, 0,  invs + 1 * 4096, x1, 64, invp + 1 * 4096, out, wo);
    ortho_kernel<<<16, 256, 0, stream>>>(x3, 0,  invs + 3 * 4096, x3, 64, invp + 3 * 4096, out, wo);
    // private-vs-private per view
    ortho_kernel<<<16, 256, 0, stream>>>(x0, 64, invp + 0 * 4096, x2, 64, invp + 2 * 4096, out, wo);
    ortho_kernel<<<16, 256, 0, stream>>>(x1, 64, invp + 1 * 4096, x3, 64, invp + 3 * 4096, out, wo);
}